// CompressedWaveletEmbedding_80728205296211
// MI455X (gfx1250) — compile-verified
//
#include <hip/hip_runtime.h>

typedef __attribute__((ext_vector_type(2))) float v2f;
typedef __attribute__((ext_vector_type(8))) float v8f;

#define B_   8
#define T_   4096
#define D_   1024
#define SEG  256   // input rows per block segment (closed under 8 lifting levels)
#define DC   16    // d-columns per block (= WMMA N dimension)
#define PAD  18    // LDS row pitch in floats (even -> 8B aligned rows, conflict-free)

// ---------------------------------------------------------------------------
// Kernel 1: levels 0..7. One block = (b, 256-row time segment, 16-col d chunk).
// All intermediates live in LDS; details stream to d_out; level-7 approx -> ws.
// Predict step done on the matrix pipe: detail = odd + even x Diag(-pw) via
// four V_WMMA_F32_16X16X4_F32 accumulations (exact f32).
// ---------------------------------------------------------------------------
__global__ __launch_bounds__(256) void dwt_levels0_7(
    const float* __restrict__ x,
    const float* __restrict__ pw, const float* __restrict__ uw,
    const float* __restrict__ ns, const float* __restrict__ nd,
    float* __restrict__ out, float* __restrict__ ws)
{
  __shared__ float bufA[SEG * PAD];        // holds level inputs (even lvl src)
  __shared__ float bufB[(SEG / 2) * PAD];  // ping-pong partner

  const int tid   = threadIdx.x;
  const int blk   = blockIdx.x;          // B_*16*64 = 8192 blocks
  const int b     = blk >> 10;           // / (16 segs * 64 d-chunks)
  const int tseg  = (blk >> 6) & 15;
  const int dchk  = blk & 63;
  const int R0    = tseg * SEG;          // first input row of this segment
  const int dbase = dchk * DC;

  // Cooperative load: 256 rows x 16 cols (64B contiguous per row chunk).
  {
    const float* src = x + ((size_t)b * T_ + R0) * D_ + dbase;
#pragma unroll
    for (int i = 0; i < (SEG * DC) / 256; ++i) {
      int idx = tid + i * 256;
      int r = idx >> 4, c = idx & 15;
      bufA[r * PAD + c] = src[(size_t)r * D_ + c];
    }
  }
  __syncthreads();

  const int w    = tid >> 5;   // wave id (wave32)
  const int lane = tid & 31;
  const int h    = lane >> 4;  // lane half
  const int n    = lane & 15;  // N index (local d) / M index for A operand
  const int gd   = dbase + n;  // global d column for this lane (WMMA path)

  for (int lvl = 0; lvl < 8; ++lvl) {
    float* srcB = (lvl & 1) ? bufB : bufA;
    float* dstB = (lvl & 1) ? bufA : bufB;
    const int Lp   = 128 >> lvl;        // pairs at this level within the block
    const int gpo  = R0 >> (lvl + 1);   // this block's global pair offset
    const int tofs = T_ >> (lvl + 1);   // time offset of detail_lvl in output

    if (Lp >= 16) {
      // ---- WMMA tile path: wave w handles pairs [16w, 16w+16) ----
      if (w * 16 < Lp) {
        const int prow = w * 16;
        const float wpn = -pw[lvl * D_ + gd];
        const float wuv =  uw[lvl * D_ + gd];
        const float wsv =  ns[lvl * D_ + gd];
        const float wdv =  nd[lvl * D_ + gd];

        // C/D layout: lane(h,n), vgpr j <-> element [p = prow+j+8h, d = n]
        v8f acc, ev;
#pragma unroll
        for (int j = 0; j < 8; ++j) {
          int p  = prow + j + 8 * h;
          acc[j] = srcB[(2 * p + 1) * PAD + n];  // odd  -> C accumulator
          ev[j]  = srcB[(2 * p    ) * PAD + n];  // even (for update step)
        }

        // detail = odd + sum_c even[:,4c..4c+3] x DiagChunk_c(-pw)
        // A layout: lane(h,n) v0/v1 = A[M=n, K=2h]/A[M=n, K=2h+1]
        // B layout: lane(h,n) v0/v1 = B[K=2h, N=n]/B[K=2h+1, N=n]
        const int arow = 2 * (prow + n);
#pragma unroll
        for (int c = 0; c < 4; ++c) {
          v2f Am, Bm;
          Am.x = srcB[arow * PAD + 4 * c + 2 * h];
          Am.y = srcB[arow * PAD + 4 * c + 2 * h + 1];
          Bm.x = (n == 4 * c + 2 * h    ) ? wpn : 0.0f;
          Bm.y = (n == 4 * c + 2 * h + 1) ? wpn : 0.0f;
          acc = __builtin_amdgcn_wmma_f32_16x16x4_f32(
              false, Am, false, Bm, (short)0, acc, false, false);
        }

        // acc = detail. Scaled detail -> global; approx -> dst LDS buffer.
#pragma unroll
        for (int j = 0; j < 8; ++j) {
          int p = prow + j + 8 * h;
          out[((size_t)b * T_ + tofs + gpo + p) * D_ + gd] = acc[j] * wdv;
          dstB[p * PAD + n] = fmaf(wuv, acc[j], ev[j]) * wsv;
        }
      }
    } else {
      // ---- small-level VALU path (Lp = 8,4,2,1) ----
      int items = Lp * DC;
      if (tid < items) {
        int p = tid >> 4, c = tid & 15;
        int col = dbase + c;
        float evs = srcB[(2 * p) * PAD + c];
        float ods = srcB[(2 * p + 1) * PAD + c];
        float det = ods - pw[lvl * D_ + col] * evs;
        float ap  = (evs + uw[lvl * D_ + col] * det) * ns[lvl * D_ + col];
        out[((size_t)b * T_ + tofs + gpo + p) * D_ + col] = det * nd[lvl * D_ + col];
        if (lvl == 7)
          ws[((size_t)b * 16 + tseg) * D_ + col] = ap;   // level-7 approx row
        else
          dstB[p * PAD + c] = ap;
      }
    }
    __syncthreads();
  }
}

// ---------------------------------------------------------------------------
// Kernel 2: levels 8..11 on the 16-row approx tensor. One thread = one
// (b, d) column held fully in registers; writes detail_8..11 and final approx.
// ---------------------------------------------------------------------------
__global__ __launch_bounds__(256) void dwt_levels8_11(
    const float* __restrict__ ws,
    const float* __restrict__ pw, const float* __restrict__ uw,
    const float* __restrict__ ns, const float* __restrict__ nd,
    float* __restrict__ out)
{
  int gid = blockIdx.x * 256 + threadIdx.x;  // B_*D_ = 8192 threads
  int b = gid >> 10;
  int d = gid & (D_ - 1);

  float a[16];
#pragma unroll
  for (int r = 0; r < 16; ++r) a[r] = ws[((size_t)b * 16 + r) * D_ + d];

#pragma unroll
  for (int l = 0; l < 4; ++l) {
    const int lvl = 8 + l;
    const int L = 8 >> l;  // pairs at this level
    const float pwv = pw[lvl * D_ + d], uwv = uw[lvl * D_ + d];
    const float nsv = ns[lvl * D_ + d], ndv = nd[lvl * D_ + d];
#pragma unroll
    for (int p = 0; p < 8; ++p) {
      if (p < L) {
        float evs = a[2 * p], ods = a[2 * p + 1];
        float det = ods - pwv * evs;
        float ap  = (evs + uwv * det) * nsv;
        out[((size_t)b * T_ + L + p) * D_ + d] = det * ndv;
        a[p] = ap;
      }
    }
  }
  out[((size_t)b * T_) * D_ + d] = a[0];  // final approx at t = 0
}

// ---------------------------------------------------------------------------
extern "C" void kernel_launch(void* const* d_in, const int* in_sizes, int n_in,
                              void* d_out, int out_size, void* d_ws, size_t ws_size,
                              hipStream_t stream) {
  (void)in_sizes; (void)n_in; (void)out_size; (void)ws_size;
  const float* x  = (const float*)d_in[0];
  const float* pw = (const float*)d_in[1];
  const float* uw = (const float*)d_in[2];
  const float* ns = (const float*)d_in[3];
  const float* nd = (const float*)d_in[4];
  float* out = (float*)d_out;
  float* ws  = (float*)d_ws;  // needs only 8*16*1024*4 = 512 KB scratch

  dwt_levels0_7 <<<B_ * 16 * 64, 256, 0, stream>>>(x, pw, uw, ns, nd, out, ws);
  dwt_levels8_11<<<(B_ * D_) / 256, 256, 0, stream>>>(ws, pw, uw, ns, nd, out);
}